// ArgmaxBase_46523085750826
// MI455X (gfx1250) — compile-verified
//
#include <hip/hip_runtime.h>
#include <hip/hip_bf16.h>
#include <stdint.h>

// ---------------------------------------------------------------------------
// Dequantization-noise rescale (uniform dequantizer), MI455X / gfx1250.
// Memory-bound (~463 MB total traffic -> ~20us at 23.3 TB/s). No matrix ops
// exist in this workload, so no WMMA; the CDNA5-specific path used is the
// async global<->LDS DMA (ASYNCcnt-tracked) for the pure-copy half of the
// traffic, which moves 256 MB without touching the VGPR file.
// ---------------------------------------------------------------------------

#define B_ROWS   524288
#define C_DIMS   64
#define D_DIMS   8
#define NB_BITS  21
#define ROW_OUT  106            // C_DIMS + 2*NB_BITS

typedef __attribute__((address_space(3))) char lds_char;

// ---------------------------------------------------------------------------
// Kernel 1: out[:, 0:64] = inputs_continuous, via async global->LDS->global.
// One wave handles one full row: lane c moves the c-th float2 of the row, so
// each async instruction transfers a contiguous 256B chunk per wave on both
// the load and the store side. b64 granularity because the output row stride
// (424B) is only 8-byte aligned on odd rows.
// ---------------------------------------------------------------------------
__global__ void copy_continuous_async(const float* __restrict__ ic,
                                      float* __restrict__ out) {
    __shared__ char smem[256 * 8];
    unsigned g = blockIdx.x * blockDim.x + threadIdx.x;
    unsigned r = g >> 5;        // row index
    unsigned c = g & 31;        // float2 column within the row (64 floats = 32 x float2)

    const float2* src = (const float2*)ic + ((size_t)r * 32u + c);
    float2*       dst = (float2*)out      + ((size_t)r * 53u + c);   // 106/2 = 53

    // LDS staging slot for this lane (8B, 8B-aligned). AS3 pointer = 32-bit
    // LDS offset in one VGPR, exactly what the async ops want for dsaddr.
    lds_char* lp = (lds_char*)(&smem[threadIdx.x * 8]);

    // global -> LDS (ASYNCcnt++), no VGPR data path
    asm volatile("global_load_async_to_lds_b64 %0, %1, off"
                 :: "v"(lp), "v"(src) : "memory");
    // data resident in LDS once ASYNCcnt drains
    asm volatile("s_wait_asynccnt 0" ::: "memory");
    // LDS -> global (ASYNCcnt++); S_ENDPGM's implicit wait-idle covers completion
    asm volatile("global_store_async_from_lds_b64 %0, %1, off"
                 :: "v"(dst), "v"(lp) : "memory");
}

// ---------------------------------------------------------------------------
// Kernel 2: per-row mixed-radix encode + conditional swap/scale of the noise
// pairs, writing out[:, 64:106] and logp[r] = sum(log(max_v)).
// (reference: result = 0 - rescaling_log_probs = +sum(log(max_v)))
// ---------------------------------------------------------------------------
__global__ void noise_rescale_kernel(const int*   __restrict__ cat,
                                     const float* __restrict__ noise,
                                     const int*   __restrict__ factors,
                                     const int*   __restrict__ mask,
                                     float* __restrict__ out) {
    unsigned r = blockIdx.x * blockDim.x + threadIdx.x;

    // 8 categorical ints per row = 32B, read as two int4 (wave reads a
    // contiguous 1KB block). factors[] is uniform + constant-indexed -> SMEM.
    const int4* cat4 = (const int4*)cat + (size_t)r * 2u;
    int4 a = cat4[0];
    int4 b = cat4[1];
    int code = factors[0] * a.x + factors[1] * a.y + factors[2] * a.z + factors[3] * a.w
             + factors[4] * b.x + factors[5] * b.y + factors[6] * b.z + factors[7] * b.w;

    const float2* nz   = (const float2*)noise + (size_t)r * NB_BITS;       // 168B row, 8B aligned
    float2*       orow = (float2*)(out + (size_t)r * ROW_OUT + C_DIMS);    // 8B aligned

    float acc = 0.0f;
#pragma unroll
    for (int i = 0; i < NB_BITS; ++i) {
        float2 v  = nz[i];
        bool bit  = (code & mask[i]) != 0;       // mask[] uniform -> SMEM
        float mx  = bit ? v.x : v.y;             // gather "max" slot
        float mn  = bit ? v.y : v.x;             // gather "min" slot
        acc      += __logf(mx);
        float nm  = mn * mx;                     // rescaled min slot
        float2 o;
        o.x = bit ? mx : nm;                     // scatter back
        o.y = bit ? nm : mx;
        orow[i] = o;
    }

    // second output: dequant_log_probs(0) - rescaling_log_probs(-acc) = acc
    float* logp = out + (size_t)B_ROWS * ROW_OUT;
    logp[r] = acc;
}

extern "C" void kernel_launch(void* const* d_in, const int* in_sizes, int n_in,
                              void* d_out, int out_size, void* d_ws, size_t ws_size,
                              hipStream_t stream) {
    (void)in_sizes; (void)n_in; (void)out_size; (void)d_ws; (void)ws_size;

    const float* ic  = (const float*)d_in[0];   // [B, 64]  f32
    const int*   cat = (const int*)  d_in[1];   // [B, 8]   i32
    const float* nz  = (const float*)d_in[2];   // [B, 42]  f32
    const int*   fac = (const int*)  d_in[3];   // [8]      i32
    const int*   msk = (const int*)  d_in[4];   // [21]     i32
    float* out = (float*)d_out;                 // [B*106] ++ [B]

    dim3 blk(256);
    // B*32 lanes, one float2 each: 65536 blocks
    copy_continuous_async<<<(B_ROWS * 32) / 256, blk, 0, stream>>>(ic, out);
    // one lane per row: 2048 blocks
    noise_rescale_kernel<<<B_ROWS / 256, blk, 0, stream>>>(cat, nz, fac, msk, out);
}